// CNNNer_44169443672535
// MI455X (gfx1250) — compile-verified
//
#include <hip/hip_runtime.h>
#include <hip/hip_bf16.h>
#include <math.h>

// Problem constants
#define B_    2
#define N_    1024
#define HID_  768
#define BSZ_  200
#define WHALF 64
#define WFULL 129          // 2*WHALF+1
#define TAGS  9
#define JP    208          // BSZ+1 padded to mult of 16
#define NC    1824         // 201*9=1809 padded to mult of 16 (114 tiles)
#define BN    (B_*N_)      // 2048

typedef __attribute__((ext_vector_type(2))) float v2f;
typedef __attribute__((ext_vector_type(8))) float v8f;

#define WMMA_F32(A, B, C) \
    __builtin_amdgcn_wmma_f32_16x16x4_f32(false, (A), false, (B), (short)0, (C), false, false)

// ---------------------------------------------------------------------------
// Gp precompute: Gp[i, j*9+t] = sum_k U[k,i,j]*Wd[k,t]  (+ folded Wcat / bd)
// ---------------------------------------------------------------------------
__global__ void gzero_kernel(float* __restrict__ Gp) {
    int idx = blockIdx.x * blockDim.x + threadIdx.x;
    if (idx < JP * NC) Gp[idx] = 0.0f;
}

__global__ void gcomp_kernel(const float* __restrict__ U,
                             const float* __restrict__ Wcat,
                             const float* __restrict__ Wd,
                             const float* __restrict__ bd,
                             float* __restrict__ Gp) {
    int idx = blockIdx.x * blockDim.x + threadIdx.x;   // over 201*201 (i,j)
    if (idx >= 201 * 201) return;
    int i = idx / 201;
    int j = idx - i * 201;
    float acc[TAGS];
#pragma unroll
    for (int t = 0; t < TAGS; ++t) acc[t] = 0.0f;
    for (int k = 0; k < BSZ_; ++k) {
        float w = U[(size_t)k * (201 * 201) + idx];
        if (i == 200) w += Wcat[(size_t)k * 402 + 201 + j];   // Q fold (Wtp^T Wd)
        if (j == 200) w += Wcat[(size_t)k * 402 + i];         // P fold (Whp^T Wd)
        const float* wd = Wd + (size_t)k * TAGS;
#pragma unroll
        for (int t = 0; t < TAGS; ++t) acc[t] += w * wd[t];
    }
    if (i == 200 && j == 200) {
#pragma unroll
        for (int t = 0; t < TAGS; ++t) acc[t] += bd[t];
    }
    float* g = Gp + (size_t)i * NC + (size_t)j * TAGS;
#pragma unroll
    for (int t = 0; t < TAGS; ++t) g[t] = acc[t];
}

// ---------------------------------------------------------------------------
// head/tail pad columns: col 200 = 1.0 (ones concat), cols 201..207 = 0
// ---------------------------------------------------------------------------
__global__ void pad_kernel(float* __restrict__ headp, float* __restrict__ tailp) {
    int idx = blockIdx.x * blockDim.x + threadIdx.x;   // BN * 8
    if (idx >= BN * 8) return;
    int m = idx >> 3;
    int c = 200 + (idx & 7);
    float v = (c == 200) ? 1.0f : 0.0f;
    headp[(size_t)m * JP + c] = v;
    tailp[(size_t)m * JP + c] = v;
}

__device__ __forceinline__ float gelu_exact(float x) {
    return 0.5f * x * (1.0f + erff(x * 0.70710678f));
}

// ---------------------------------------------------------------------------
// head/tail GEMM + exact GELU via fp32 WMMA, 2(M)x1(N) tiles per wave.
// C(2048 x 400) = state(2048 x 768) @ [Wh | Wt](768 x 400)
// ---------------------------------------------------------------------------
__global__ void ht_kernel(const float* __restrict__ state,
                          const float* __restrict__ Wh, const float* __restrict__ bh,
                          const float* __restrict__ Wt, const float* __restrict__ bt,
                          float* __restrict__ headp, float* __restrict__ tailp) {
    const int NT = 400 / 16;                              // 25 col tiles
    const int MP = BN / 32;                               // 64 row-pair tiles
    int wave = blockIdx.x * (blockDim.x >> 5) + (threadIdx.x >> 5);
    if (wave >= MP * NT) return;
    int tm = wave / NT, tn = wave - tm * NT;
    int lane = threadIdx.x & 31;
    int half = lane >> 4;                                 // K-pair select
    int lr   = lane & 15;                                 // A row / B,D col
    int m0 = tm * 32, n0 = tn * 16;

    int col = n0 + lr;
    const float* Bmat = (col < BSZ_) ? Wh : Wt;
    int cc = (col < BSZ_) ? col : col - BSZ_;
    const float* arow0 = state + (size_t)(m0 + lr) * HID_;
    const float* arow1 = arow0 + (size_t)16 * HID_;

    v8f acc0 = {}, acc1 = {};
    for (int k = 0; k < HID_; k += 4) {
        int ka = k + half * 2;
        v2f a0, a1, b;
        a0.x = arow0[ka];  a0.y = arow0[ka + 1];
        a1.x = arow1[ka];  a1.y = arow1[ka + 1];
        b.x  = Bmat[(size_t)ka * BSZ_ + cc];
        b.y  = Bmat[(size_t)(ka + 1) * BSZ_ + cc];
        acc0 = WMMA_F32(a0, b, acc0);
        acc1 = WMMA_F32(a1, b, acc1);
    }
    float bias = (col < BSZ_) ? bh[cc] : bt[cc];
    float* outb = (col < BSZ_) ? headp : tailp;
#pragma unroll
    for (int r = 0; r < 8; ++r) {
        int m = m0 + half * 8 + r;                        // D layout: VGPR r -> M=r / M=8+r
        outb[(size_t)m * JP + cc]        = gelu_exact(acc0[r] + bias);
        outb[(size_t)(m + 16) * JP + cc] = gelu_exact(acc1[r] + bias);
    }
}

// ---------------------------------------------------------------------------
// V GEMM via fp32 WMMA, 2(M)x2(N) tiles per wave (8 FLOP/byte from L1).
// V(2048 x 1824) = headp(2048 x 208) @ Gp(208 x 1824)
// ---------------------------------------------------------------------------
__global__ void v_kernel(const float* __restrict__ headp,
                         const float* __restrict__ Gp,
                         float* __restrict__ V) {
    const int NP = NC / 32;                               // 57 col-pair tiles
    const int MP = BN / 32;                               // 64 row-pair tiles
    int wave = blockIdx.x * (blockDim.x >> 5) + (threadIdx.x >> 5);
    if (wave >= MP * NP) return;
    int tm = wave / NP, tn = wave - tm * NP;
    int lane = threadIdx.x & 31;
    int half = lane >> 4;
    int lr   = lane & 15;
    int m0 = tm * 32, n0 = tn * 32;
    int col0 = n0 + lr, col1 = col0 + 16;
    const float* arow0 = headp + (size_t)(m0 + lr) * JP;
    const float* arow1 = arow0 + (size_t)16 * JP;

    v8f acc00 = {}, acc01 = {}, acc10 = {}, acc11 = {};
    for (int k = 0; k < JP; k += 4) {
        int ka = k + half * 2;
        const float* g0 = Gp + (size_t)ka * NC;
        const float* g1 = g0 + NC;
        v2f a0, a1, b0, b1;
        a0.x = arow0[ka];  a0.y = arow0[ka + 1];
        a1.x = arow1[ka];  a1.y = arow1[ka + 1];
        b0.x = g0[col0];   b0.y = g1[col0];
        b1.x = g0[col1];   b1.y = g1[col1];
        acc00 = WMMA_F32(a0, b0, acc00);
        acc01 = WMMA_F32(a0, b1, acc01);
        acc10 = WMMA_F32(a1, b0, acc10);
        acc11 = WMMA_F32(a1, b1, acc11);
    }
#pragma unroll
    for (int r = 0; r < 8; ++r) {
        int m = m0 + half * 8 + r;
        V[(size_t)m * NC + col0]        = acc00[r];
        V[(size_t)m * NC + col1]        = acc01[r];
        V[(size_t)(m + 16) * NC + col0] = acc10[r];
        V[(size_t)(m + 16) * NC + col1] = acc11[r];
    }
}

// ---------------------------------------------------------------------------
// Band scoring: out[b,n,r,t] = masked( sum_j tailp[b, n+r-64, j] * V[bn, j*9+t] )
// (h2@Wd, t2@Wd, bd already folded into V through Gp + ones columns)
// ---------------------------------------------------------------------------
__global__ void s_kernel(const float* __restrict__ tailp,
                         const float* __restrict__ V,
                         const int* __restrict__ lengths,
                         const float* __restrict__ bd,
                         float* __restrict__ out) {
    __shared__ float Vs[1809];
    __shared__ float bds[TAGS];
    int bn = blockIdx.x;                                   // one (b,n) per block
    int b = bn >> 10, n = bn & (N_ - 1);
    for (int i = threadIdx.x; i < 1809; i += blockDim.x)
        Vs[i] = V[(size_t)bn * NC + i];
    if (threadIdx.x < TAGS) bds[threadIdx.x] = bd[threadIdx.x];
    __syncthreads();

    int len = lengths[b];
    bool q_ok = (n < len);
    float* obase = out + (size_t)bn * (WFULL * TAGS);
    for (int idx = threadIdx.x; idx < WFULL * TAGS; idx += blockDim.x) {
        int r = idx / TAGS;
        int t = idx - r * TAGS;
        int j = n + r - WHALF;
        float s;
        if (q_ok && j >= 0 && j < N_ && j < len) {
            const float* trow = tailp + (size_t)(b * N_ + j) * JP;
            float acc = 0.0f;
            for (int jj = 0; jj < 201; ++jj)
                acc = fmaf(trow[jj], Vs[jj * TAGS + t], acc);
            s = acc;
        } else {
            s = bds[t];                                    // u==0 path: scores = bd
        }
        if (!isfinite(s)) s = 0.0f;
        obase[idx] = s;
    }
}

// ---------------------------------------------------------------------------
extern "C" void kernel_launch(void* const* d_in, const int* in_sizes, int n_in,
                              void* d_out, int out_size, void* d_ws, size_t ws_size,
                              hipStream_t stream) {
    const float* state = (const float*)d_in[0];
    const int*   lengths = (const int*)d_in[1];
    const float* Wh  = (const float*)d_in[2];
    const float* bh  = (const float*)d_in[3];
    const float* Wt  = (const float*)d_in[4];
    const float* bt  = (const float*)d_in[5];
    const float* U   = (const float*)d_in[6];
    const float* Wcat= (const float*)d_in[7];
    const float* Wd  = (const float*)d_in[8];
    const float* bd  = (const float*)d_in[9];
    float* out = (float*)d_out;

    // workspace layout (~20 MB total)
    float* headp = (float*)d_ws;                  // BN x JP
    float* tailp = headp + (size_t)BN * JP;       // BN x JP
    float* Gp    = tailp + (size_t)BN * JP;       // JP x NC
    float* V     = Gp    + (size_t)JP * NC;       // BN x NC

    // Gp = U x_k Wd (+ folds)
    gzero_kernel<<<(JP * NC + 255) / 256, 256, 0, stream>>>(Gp);
    gcomp_kernel<<<(201 * 201 + 255) / 256, 256, 0, stream>>>(U, Wcat, Wd, bd, Gp);

    // head/tail projections (WMMA f32) + pad/ones columns
    pad_kernel<<<(BN * 8 + 255) / 256, 256, 0, stream>>>(headp, tailp);
    {
        int waves = (BN / 32) * (400 / 16);       // 1600
        ht_kernel<<<waves / 8, 256, 0, stream>>>(state, Wh, bh, Wt, bt, headp, tailp);
    }

    // V = headp @ Gp (WMMA f32, 2x2 register blocking)
    {
        int waves = (BN / 32) * (NC / 32);        // 3648
        v_kernel<<<waves / 8, 256, 0, stream>>>(headp, Gp, V);
    }

    // band scores + masking
    s_kernel<<<BN, 256, 0, stream>>>(tailp, V, lengths, bd, out);
}